// GCNGraphClassifier_2156073582828
// MI455X (gfx1250) — compile-verified
//
#include <hip/hip_runtime.h>
#include <math.h>

#define NNODES 100000
#define NEDGES 3200000
#define NGRAPH 1024
#define FIN    4
#define FH     32
#define FOUT   1317
#define REDB   240   // grid-stride reduction blocks

typedef __attribute__((ext_vector_type(2))) float v2f;
typedef __attribute__((ext_vector_type(8))) float v8f;

// ---------------- utility ----------------
__global__ void k_zero(float* p, int n) {
  int t = blockIdx.x * blockDim.x + threadIdx.x;
  if (t < n) p[t] = 0.f;
}

// ---------------- GCN normalization ----------------
__global__ void k_deg(const int* __restrict__ col, const float* __restrict__ ea,
                      float* __restrict__ deg, int E) {
  int e = blockIdx.x * blockDim.x + threadIdx.x;
  if (e < E) atomicAdd(&deg[col[e]], ea[e]);
}

__global__ void k_dinv(const float* __restrict__ deg, float* __restrict__ dinv, int n) {
  int i = blockIdx.x * blockDim.x + threadIdx.x;
  if (i < n) dinv[i] = rsqrtf(deg[i] + 1.0f);   // deg+self-loop >= 1 always
}

__global__ void k_norm(const int* __restrict__ row, const int* __restrict__ col,
                       const float* __restrict__ ea, const float* __restrict__ dinv,
                       float* __restrict__ norm, int E) {
  int e = blockIdx.x * blockDim.x + threadIdx.x;
  if (e < E) norm[e] = dinv[row[e]] * ea[e] * dinv[col[e]];
}

// ---------------- f32 WMMA GEMM: out[M x n_cols] = A[M x K] * W[n_cols x K]^T (+bias) ----
// One wave computes one 16x16 tile via K/4 chained V_WMMA_F32_16X16X4_F32.
// A frag (16x4): lane L holds m=L&15, k = reg + 2*(L>>4)   -> contiguous float2
// B frag (4x16): lane L holds n=L&15, k = reg + 2*(L>>4)   -> contiguous float2 of W row
// C/D (16x16):  lane L reg r holds m = r + 8*(L>>4), n = L&15
//
// Out-of-range output columns: clamp the W row index to 0 and load anyway.
// Column n of D lives only in lanes with (lane&15)==n, which are exactly the
// lanes that skip the store, so the garbage accumulation is dead. This keeps
// EXEC untouched in the unrolled WMMA chain (no saveexec per K-step).
// K and LDC are compile-time so loads unroll and the 8 row stores fold their
// strides into the global_store immediate-offset field.
template <int K, int LDC>
__global__ void k_wmma_gemm_nt(const float* __restrict__ A,
                               const float* __restrict__ W,
                               const float* __restrict__ bias,
                               float* __restrict__ out,
                               int n_cols, int row_tiles, int col_tiles) {
  int wave = (blockIdx.x * blockDim.x + threadIdx.x) >> 5;
  if (wave >= row_tiles * col_tiles) return;      // wave-uniform exit
  int lane = threadIdx.x & 31;
  int rt = wave / col_tiles;
  int ct = wave - rt * col_tiles;
  int m  = lane & 15;
  int hi = lane >> 4;
  int arow = rt * 16 + m;        // A row for this lane
  int jb   = ct * 16 + m;        // B/D column for this lane
  bool jok = jb < n_cols;
  int jcl  = jok ? jb : 0;       // clamped, always-valid W row
  const float* ap = A + (size_t)arow * K + 2 * hi;
  const float* bp = W + (size_t)jcl * K + 2 * hi;

  v8f c = {0.f, 0.f, 0.f, 0.f, 0.f, 0.f, 0.f, 0.f};
#pragma unroll
  for (int kk = 0; kk < K; kk += 4) {
    v2f a = *(const v2f*)(ap + kk);
    v2f b = *(const v2f*)(bp + kk);
    c = __builtin_amdgcn_wmma_f32_16x16x4_f32(false, a, false, b,
                                              (short)0, c, false, false);
  }
  if (jok) {
    float bv = bias ? bias[jb] : 0.f;
    int mbase = rt * 16 + 8 * hi;
    float* op = out + (size_t)mbase * LDC + jb;   // one base address, 8 imm-offset stores
#pragma unroll
    for (int r = 0; r < 8; ++r)
      op[(size_t)r * LDC] = c[r] + bv;
  }
}

// ---------------- sparse scatter: agg[col] += norm * hw[row], feature-per-lane ----------
__global__ void k_scatter(const int* __restrict__ row, const int* __restrict__ col,
                          const float* __restrict__ norm, const float* __restrict__ hw,
                          float* __restrict__ agg, int E) {
  int t = blockIdx.x * blockDim.x + threadIdx.x;
  int e = t >> 5;
  if (e >= E) return;
  int j = t & 31;
  float w = norm[e];
  int r = row[e], c = col[e];
  atomicAdd(&agg[(size_t)c * FH + j], w * hw[(size_t)r * FH + j]);
}

// h = relu(agg + dinv^2 * hw + b)
__global__ void k_finalize(const float* __restrict__ agg, const float* __restrict__ hw,
                           const float* __restrict__ dinv, const float* __restrict__ bias,
                           float* __restrict__ h, int n) {
  int t = blockIdx.x * blockDim.x + threadIdx.x;
  if (t >= n * FH) return;
  int node = t >> 5, j = t & 31;
  float sn = dinv[node];
  sn *= sn;
  float v = agg[t] + sn * hw[t] + bias[j];
  h[t] = fmaxf(v, 0.f);
}

// ---------------- attention ----------------
__global__ void k_score(const float* __restrict__ h, const float* __restrict__ Wa,
                        const float* __restrict__ ba, float* __restrict__ s, int n) {
  int t = blockIdx.x * blockDim.x + threadIdx.x;
  if (t >= n * FH) return;
  int node = t >> 5, j = t & 31;
  float v = h[t] * Wa[j];
#pragma unroll
  for (int off = 16; off; off >>= 1) v += __shfl_xor(v, off, 32);
  if (j == 0) s[node] = v + ba[0];
}

__global__ void k_reduce_max(const float* __restrict__ x, int n, float* __restrict__ partial) {
  __shared__ float sm[256];
  float m = -3.402823466e38f;
  for (int i = blockIdx.x * blockDim.x + threadIdx.x; i < n; i += gridDim.x * blockDim.x)
    m = fmaxf(m, x[i]);
  sm[threadIdx.x] = m;
  __syncthreads();
  for (int o = 128; o; o >>= 1) {
    if ((int)threadIdx.x < o) sm[threadIdx.x] = fmaxf(sm[threadIdx.x], sm[threadIdx.x + o]);
    __syncthreads();
  }
  if (threadIdx.x == 0) partial[blockIdx.x] = sm[0];
}

__global__ void k_reduce_sum(const float* __restrict__ x, int n, float* __restrict__ partial) {
  __shared__ float sm[256];
  float a = 0.f;
  for (int i = blockIdx.x * blockDim.x + threadIdx.x; i < n; i += gridDim.x * blockDim.x)
    a += x[i];
  sm[threadIdx.x] = a;
  __syncthreads();
  for (int o = 128; o; o >>= 1) {
    if ((int)threadIdx.x < o) sm[threadIdx.x] += sm[threadIdx.x + o];
    __syncthreads();
  }
  if (threadIdx.x == 0) partial[blockIdx.x] = sm[0];
}

__global__ void k_exp_sum(const float* __restrict__ s, int n, const float* __restrict__ Mptr,
                          float* __restrict__ e_out, float* __restrict__ partial) {
  __shared__ float sm[256];
  float M = Mptr[0];
  float acc = 0.f;
  for (int i = blockIdx.x * blockDim.x + threadIdx.x; i < n; i += gridDim.x * blockDim.x) {
    float v = expf(s[i] - M);
    e_out[i] = v;
    acc += v;
  }
  sm[threadIdx.x] = acc;
  __syncthreads();
  for (int o = 128; o; o >>= 1) {
    if ((int)threadIdx.x < o) sm[threadIdx.x] += sm[threadIdx.x + o];
    __syncthreads();
  }
  if (threadIdx.x == 0) partial[blockIdx.x] = sm[0];
}

__global__ void k_scale(float* __restrict__ a, const float* __restrict__ Sptr, int n) {
  int t = blockIdx.x * blockDim.x + threadIdx.x;
  if (t < n) a[t] /= Sptr[0];
}

// x_weighted[j] = sum_n h[n,j]*attn[n], hierarchical (LDS -> 32 global atomics/block)
__global__ void k_xweighted(const float* __restrict__ h, const float* __restrict__ attn,
                            float* __restrict__ xw, int n) {
  __shared__ float loc[FH];
  if (threadIdx.x < FH) loc[threadIdx.x] = 0.f;
  __syncthreads();
  for (int t = blockIdx.x * blockDim.x + threadIdx.x; t < n * FH; t += gridDim.x * blockDim.x)
    atomicAdd(&loc[t & 31], h[t] * attn[t >> 5]);
  __syncthreads();
  if (threadIdx.x < FH) atomicAdd(&xw[threadIdx.x], loc[threadIdx.x]);
}

// ---------------- mean pooling ----------------
__global__ void k_pool(const float* __restrict__ h, const int* __restrict__ batch,
                       float* __restrict__ pooled, float* __restrict__ cnt, int n) {
  int t = blockIdx.x * blockDim.x + threadIdx.x;
  if (t >= n * FH) return;
  int node = t >> 5, j = t & 31;
  int g = batch[node];
  atomicAdd(&pooled[(size_t)g * FH + j], h[t]);
  if (j == 0) atomicAdd(&cnt[g], 1.0f);
}

__global__ void k_pool_div(float* __restrict__ pooled, const float* __restrict__ cnt, int n) {
  int t = blockIdx.x * blockDim.x + threadIdx.x;
  if (t < n) pooled[t] /= fmaxf(cnt[t >> 5], 1.0f);
}

// ---------------- per-row log_softmax (in place) ----------------
__global__ void k_logsoftmax(float* __restrict__ logits, int ncols) {
  __shared__ float sm[256];
  float* rowp = logits + (size_t)blockIdx.x * ncols;
  float m = -3.402823466e38f;
  for (int i = threadIdx.x; i < ncols; i += blockDim.x) m = fmaxf(m, rowp[i]);
  sm[threadIdx.x] = m;
  __syncthreads();
  for (int o = 128; o; o >>= 1) {
    if ((int)threadIdx.x < o) sm[threadIdx.x] = fmaxf(sm[threadIdx.x], sm[threadIdx.x + o]);
    __syncthreads();
  }
  float M = sm[0];
  __syncthreads();
  float acc = 0.f;
  for (int i = threadIdx.x; i < ncols; i += blockDim.x) acc += expf(rowp[i] - M);
  sm[threadIdx.x] = acc;
  __syncthreads();
  for (int o = 128; o; o >>= 1) {
    if ((int)threadIdx.x < o) sm[threadIdx.x] += sm[threadIdx.x + o];
    __syncthreads();
  }
  float lse = M + logf(sm[0]);
  __syncthreads();
  for (int i = threadIdx.x; i < ncols; i += blockDim.x) rowp[i] -= lse;
}

// ---------------- host launcher ----------------
extern "C" void kernel_launch(void* const* d_in, const int* in_sizes, int n_in,
                              void* d_out, int out_size, void* d_ws, size_t ws_size,
                              hipStream_t stream) {
  (void)in_sizes; (void)n_in; (void)out_size; (void)ws_size;
  const float* x    = (const float*)d_in[0];
  const int*   eidx = (const int*)d_in[1];
  const float* ea   = (const float*)d_in[2];
  const int*   batch= (const int*)d_in[3];
  const float* W1 = (const float*)d_in[4];  const float* b1 = (const float*)d_in[5];
  const float* W2 = (const float*)d_in[6];  const float* b2 = (const float*)d_in[7];
  const float* W3 = (const float*)d_in[8];  const float* b3 = (const float*)d_in[9];
  const float* Wa = (const float*)d_in[10]; const float* ba = (const float*)d_in[11];
  const float* Wfc= (const float*)d_in[12]; const float* bfc= (const float*)d_in[13];

  const int* erow = eidx;            // edge_index[0]
  const int* ecol = eidx + NEDGES;   // edge_index[1]

  float* deg    = (float*)d_ws;
  float* dinv   = deg + NNODES;
  float* norm   = dinv + NNODES;
  float* h      = norm + NEDGES;                 // [N,32]
  float* hw     = h + (size_t)NNODES * FH;       // [N,32]
  float* agg    = hw + (size_t)NNODES * FH;      // [N,32]
  float* sc     = agg + (size_t)NNODES * FH;     // [N]
  float* pooled = sc + NNODES;                   // [G,32]
  float* cnt    = pooled + (size_t)NGRAPH * FH;  // [G]
  float* pmax   = cnt + NGRAPH;                  // [REDB]
  float* psum   = pmax + 256;                    // [REDB]
  float* scal   = psum + 256;                    // [0]=max, [1]=sum

  float* out_logits = (float*)d_out;                          // [G,OUT]
  float* out_attn   = out_logits + (size_t)NGRAPH * FOUT;     // [N]
  float* out_xw     = out_attn + NNODES;                      // [32]

  const int B = 256;
  const int NH  = NNODES * FH;
  const long long E32 = (long long)NEDGES * 32;
  auto blk = [](long long n) { return (unsigned)((n + 255) / 256); };

  // normalization coefficients
  k_zero<<<blk(NNODES), B, 0, stream>>>(deg, NNODES);
  k_deg <<<blk(NEDGES), B, 0, stream>>>(ecol, ea, deg, NEDGES);
  k_dinv<<<blk(NNODES), B, 0, stream>>>(deg, dinv, NNODES);
  k_norm<<<blk(NEDGES), B, 0, stream>>>(erow, ecol, ea, dinv, norm, NEDGES);

  const int row_tiles_n = NNODES / 16;                 // 6250
  const unsigned gemm_blocks_n = blk((long long)row_tiles_n * 2 * 32);

  // layer 1 (K = 4)
  k_wmma_gemm_nt<FIN, FH><<<gemm_blocks_n, B, 0, stream>>>(x, W1, nullptr, hw, FH, row_tiles_n, 2);
  k_zero    <<<blk(NH), B, 0, stream>>>(agg, NH);
  k_scatter <<<blk(E32), B, 0, stream>>>(erow, ecol, norm, hw, agg, NEDGES);
  k_finalize<<<blk(NH), B, 0, stream>>>(agg, hw, dinv, b1, h, NNODES);

  // layers 2 and 3 (K = 32)
  const float* Ws[2] = {W2, W3};
  const float* bs[2] = {b2, b3};
  for (int l = 0; l < 2; ++l) {
    k_wmma_gemm_nt<FH, FH><<<gemm_blocks_n, B, 0, stream>>>(h, Ws[l], nullptr, hw, FH, row_tiles_n, 2);
    k_zero    <<<blk(NH), B, 0, stream>>>(agg, NH);
    k_scatter <<<blk(E32), B, 0, stream>>>(erow, ecol, norm, hw, agg, NEDGES);
    k_finalize<<<blk(NH), B, 0, stream>>>(agg, hw, dinv, bs[l], h, NNODES);
  }

  // attention: global softmax over node scores
  k_score     <<<blk(NH), B, 0, stream>>>(h, Wa, ba, sc, NNODES);
  k_reduce_max<<<REDB, B, 0, stream>>>(sc, NNODES, pmax);
  k_reduce_max<<<1,    B, 0, stream>>>(pmax, REDB, scal);       // scal[0] = M
  k_exp_sum   <<<REDB, B, 0, stream>>>(sc, NNODES, scal, out_attn, psum);
  k_reduce_sum<<<1,    B, 0, stream>>>(psum, REDB, scal + 1);   // scal[1] = S
  k_scale     <<<blk(NNODES), B, 0, stream>>>(out_attn, scal + 1, NNODES);
  k_zero      <<<1, B, 0, stream>>>(out_xw, FH);
  k_xweighted <<<REDB, B, 0, stream>>>(h, out_attn, out_xw, NNODES);

  // mean pooling + FC + log_softmax
  k_zero    <<<blk(NGRAPH * FH + NGRAPH), B, 0, stream>>>(pooled, NGRAPH * FH + NGRAPH);
  k_pool    <<<blk(NH), B, 0, stream>>>(h, batch, pooled, cnt, NNODES);
  k_pool_div<<<blk(NGRAPH * FH), B, 0, stream>>>(pooled, cnt, NGRAPH * FH);

  const int fc_row_tiles = NGRAPH / 16;        // 64
  const int fc_col_tiles = (FOUT + 15) / 16;   // 83
  k_wmma_gemm_nt<FH, FOUT><<<blk((long long)fc_row_tiles * fc_col_tiles * 32), B, 0, stream>>>(
      pooled, Wfc, bfc, out_logits, FOUT, fc_row_tiles, fc_col_tiles);
  k_logsoftmax<<<NGRAPH, B, 0, stream>>>(out_logits, FOUT);
}